// OptNet_24257975288500
// MI455X (gfx1250) — compile-verified
//
#include <hip/hip_runtime.h>
#include <math.h>

#define NXv 64
#define NIv 128
#define NTH 256
#define NITER 5
#define INT_REG_F 1e-6f
#define INF_F __builtin_inff()

typedef __attribute__((ext_vector_type(2))) float v2f;
typedef __attribute__((ext_vector_type(8))) float v8f;
typedef __attribute__((vector_size(16))) int v4i;
typedef __attribute__((address_space(1))) v4i* p1_v4i;  // global (AS1) v4i*
typedef __attribute__((address_space(3))) v4i* p3_v4i;  // LDS (AS3) v4i*

#if defined(__gfx1250__) && __has_builtin(__builtin_amdgcn_wmma_f32_16x16x4_f32)
#define HAVE_WMMA_F32 1
#else
#define HAVE_WMMA_F32 0
#endif

#if defined(__gfx1250__) && __has_builtin(__builtin_amdgcn_global_load_async_to_lds_b128) && \
    __has_builtin(__builtin_amdgcn_s_wait_asynccnt)
#define HAVE_ASYNC_LDS 1
#else
#define HAVE_ASYNC_LDS 0
#endif

struct __align__(16) Shared {
  float U[NXv * NXv];    // Q -> U_Q (upper chol); lower triangle is scratch
  float G[NIv * NXv];    // G row-major
  float R[NIv * NIv];    // Schur complement R (kept full/symmetric)
  float S[NIv * NIv];    // per-iteration chol factor; aliased as invQ_GT (64x128) in prefactor
  float A[NXv], p[NXv], h[NIv];
  float x[NXv], rxv[NXv], t64[NXv], iqat[NXv], dxa[NXv], dxc[NXv], idU[NXv];
  float s[NIv], z[NIv], d[NIv], U12[NIv], rzv[NIv], t128[NIv], t128b[NIv], idS[NIv];
  float dsa[NIv], dza[NIv], dsc[NIv], dzc[NIv];
  float red[NTH];
  float b, u11, yv, ryv, dya, dyc, sc0, sc1;
};

// ---------- block-wide reductions ----------
__device__ __forceinline__ float block_min(Shared& sh, float a, int tid) {
  sh.red[tid] = a;
  __syncthreads();
  for (int off = NTH / 2; off > 0; off >>= 1) {
    if (tid < off) sh.red[tid] = fminf(sh.red[tid], sh.red[tid + off]);
    __syncthreads();
  }
  float r = sh.red[0];
  __syncthreads();
  return r;
}

__device__ __forceinline__ float block_sum(Shared& sh, float a, int tid) {
  sh.red[tid] = a;
  __syncthreads();
  for (int off = NTH / 2; off > 0; off >>= 1) {
    if (tid < off) sh.red[tid] += sh.red[tid + off];
    __syncthreads();
  }
  float r = sh.red[0];
  __syncthreads();
  return r;
}

// ---------- blocked upper Cholesky (panel = 16) with WMMA trailing updates ----------
// M is n x n in LDS (n multiple of 16, full square storage, symmetric input).
// Only the upper triangle of the result is meaningful for the solves.
__device__ void chol_upper_blocked(float* M, int n, int tid) {
  const int lane = tid & 31;
  const int wave = tid >> 5;
  const int hh = lane >> 4;   // lane half
  const int ll = lane & 15;   // M for A-frag / N for B,C-frags
  for (int p = 0; p < n; p += 16) {
    __syncthreads();
    // --- 1) factor 16x16 diagonal block: wave 0 only, intra-wave LDS is in-order ---
    if (wave == 0) {
      for (int k = p; k < p + 16; ++k) {
        float diag = sqrtf(M[k * n + k]);
        float dinv = 1.0f / diag;
        if (lane == 0) M[k * n + k] = diag;
        int j = k + 1 + lane;
        if (j < p + 16) {
          float ukj = M[k * n + j] * dinv;
          M[k * n + j] = ukj;
          for (int i = k + 1; i <= j; ++i) M[i * n + j] -= M[k * n + i] * ukj;
        }
      }
    }
    __syncthreads();
    const int ncols = n - p - 16;
    if (ncols <= 0) continue;
    // --- 2) panel TRSM: solve Ud^T * Panel = A, one thread per column ---
    if (tid < ncols) {
      const int c = p + 16 + tid;
      for (int i = p; i < p + 16; ++i) {
        float acc = M[i * n + c];
        for (int k = p; k < i; ++k) acc -= M[k * n + i] * M[k * n + c];
        M[i * n + c] = acc / M[i * n + i];
      }
    }
    __syncthreads();
    // --- 3) trailing update C -= Panel^T Panel via V_WMMA_F32_16X16X4_F32 ---
    const int ntile = ncols >> 4;
    const int tcnt = ntile * (ntile + 1) / 2;
    for (int t = wave; t < tcnt; t += 8) {
      int ti = 0, rem = t;
      while (rem >= ntile - ti) { rem -= (ntile - ti); ++ti; }
      const int tj = ti + rem;
      const int i0 = p + 16 + ti * 16;
      const int j0 = p + 16 + tj * 16;
#if HAVE_WMMA_F32
      v8f acc = {0.f, 0.f, 0.f, 0.f, 0.f, 0.f, 0.f, 0.f};
      for (int kk = 0; kk < 16; kk += 4) {
        v2f af, bf;
        // A[m][k] = Panel^T[m][k] = M[p+k][i0+m]; lane = m + 16h, vgpr v -> k = kk+2h+v
        af.x = M[(p + kk + 2 * hh) * n + i0 + ll];
        af.y = M[(p + kk + 2 * hh + 1) * n + i0 + ll];
        // B[k][nn] = M[p+k][j0+nn]
        bf.x = M[(p + kk + 2 * hh) * n + j0 + ll];
        bf.y = M[(p + kk + 2 * hh + 1) * n + j0 + ll];
        acc = __builtin_amdgcn_wmma_f32_16x16x4_f32(false, af, false, bf, (short)0, acc,
                                                    false, false);
      }
      for (int r = 0; r < 8; ++r) M[(i0 + r + 8 * hh) * n + j0 + ll] -= acc[r];
#else
      for (int e = lane; e < 256; e += 32) {
        int ii = i0 + (e >> 4), jj = j0 + (e & 15);
        float acc = 0.0f;
        for (int k = p; k < p + 16; ++k) acc += M[k * n + ii] * M[k * n + jj];
        M[ii * n + jj] -= acc;
      }
#endif
    }
  }
  __syncthreads();
}

// solve U^T y = v (forward), wave0 only; idg = reciprocal diagonal of U
__device__ void trsv_lt(const float* U, int n, float* v, const float* idg, int tid) {
  __syncthreads();
  if ((tid >> 5) == 0) {
    const int lane = tid & 31;
    for (int k = 0; k < n; ++k) {
      float yk = v[k] * idg[k];
      if (lane == 0) v[k] = yk;
      for (int i = k + 1 + lane; i < n; i += 32) v[i] -= U[k * n + i] * yk;
    }
  }
  __syncthreads();
}

// solve U x = v (backward), wave0 only
__device__ void trsv_ut(const float* U, int n, float* v, const float* idg, int tid) {
  __syncthreads();
  if ((tid >> 5) == 0) {
    const int lane = tid & 31;
    for (int k = n - 1; k >= 0; --k) {
      float xk = v[k] * idg[k];
      if (lane == 0) v[k] = xk;
      for (int i = lane; i < k; i += 32) v[i] -= U[i * n + k] * xk;
    }
  }
  __syncthreads();
}

// factor_kkt: S = chol_upper(R + diag(1/d + INT_REG)), then reciprocal diagonal
__device__ void factor_kkt(Shared& sh, int tid) {
  __syncthreads();
  for (int idx = tid; idx < NIv * NIv; idx += NTH) {
    int r = idx >> 7, c = idx & (NIv - 1);
    float v = sh.R[idx];
    if (r == c) v += 1.0f / sh.d[r] + INT_REG_F;
    sh.S[idx] = v;
  }
  __syncthreads();
  chol_upper_blocked(sh.S, NIv, tid);
  if (tid < NIv) sh.idS[tid] = 1.0f / sh.S[tid * NIv + tid];
  __syncthreads();
}

// solve_kkt: returns dx(64), ds(128), wz(128), wy(scalar)
__device__ void solve_kkt(Shared& sh, const float* rx, const float* rs, const float* rz,
                          float ry, float* dx, float* ds, float* wz, float* wy, int tid) {
  if (rx) {
    for (int i = tid; i < NXv; i += NTH) sh.t64[i] = rx[i];
    trsv_lt(sh.U, NXv, sh.t64, sh.idU, tid);
    trsv_ut(sh.U, NXv, sh.t64, sh.idU, tid);
  } else {
    for (int i = tid; i < NXv; i += NTH) sh.t64[i] = 0.0f;
    __syncthreads();
  }
  // H = G @ invQ_rx + rs/d - rz ; H1 = A @ invQ_rx - ry
  if (tid < NIv) {
    float acc = 0.0f;
    for (int k = 0; k < NXv; ++k) acc += sh.G[tid * NXv + k] * sh.t64[k];
    float rsv = rs ? rs[tid] : 0.0f;
    float rzl = rz ? rz[tid] : 0.0f;
    sh.t128[tid] = acc + rsv / sh.d[tid] - rzl;
  }
  if (tid == 0) {
    float acc = 0.0f;
    for (int k = 0; k < NXv; ++k) acc += sh.A[k] * sh.t64[k];
    sh.sc0 = acc - ry;
  }
  __syncthreads();
  // forward with U_S = [[u11, U12],[0, S]]
  float y0 = sh.sc0 / sh.u11;
  if (tid < NIv) sh.t128[tid] -= sh.U12[tid] * y0;
  trsv_lt(sh.S, NIv, sh.t128, sh.idS, tid);
  trsv_ut(sh.S, NIv, sh.t128, sh.idS, tid);
  if (tid == 0) {
    float acc = 0.0f;
    for (int k = 0; k < NIv; ++k) acc += sh.U12[k] * sh.t128[k];
    sh.sc1 = -((y0 - acc) / sh.u11);  // w_y
  }
  if (tid < NIv) wz[tid] = -sh.t128[tid];
  __syncthreads();
  float wyv = sh.sc1;
  if (tid == 0) *wy = wyv;
  // g1 = -rx - G^T wz - A^T wy
  if (tid < NXv) {
    float acc = -(rx ? rx[tid] : 0.0f) - sh.A[tid] * wyv;
    for (int r = 0; r < NIv; ++r) acc -= sh.G[r * NXv + tid] * wz[r];
    sh.t64[tid] = acc;
  }
  trsv_lt(sh.U, NXv, sh.t64, sh.idU, tid);
  trsv_ut(sh.U, NXv, sh.t64, sh.idU, tid);
  if (tid < NXv) dx[tid] = sh.t64[tid];
  if (tid < NIv) ds[tid] = (-(rs ? rs[tid] : 0.0f) - wz[tid]) / sh.d[tid];
  __syncthreads();
}

__device__ __forceinline__ float step_min(Shared& sh, const float* v, const float* dv, int tid) {
  float a = INF_F;
  if (tid < NIv) {
    float t = -v[tid] / dv[tid];
    a = (t > 0.0f) ? t : INF_F;  // NaN compares false -> INF, matches jnp.where
  }
  return block_min(sh, a, tid);
}

__global__ __launch_bounds__(NTH) void optnet_ipm_kernel(
    const float* __restrict__ Qg, const float* __restrict__ pg, const float* __restrict__ Ag,
    const float* __restrict__ bg, const float* __restrict__ Gg, const float* __restrict__ hg,
    float* __restrict__ out) {
  __shared__ Shared sh;
  const int tid = threadIdx.x;
  const int bix = blockIdx.x;

  const float* Qb = Qg + (size_t)bix * NXv * NXv;
  const float* pb = pg + (size_t)bix * NXv;
  const float* Ab = Ag + (size_t)bix * NXv;
  const float* Gb = Gg + (size_t)bix * NIv * NXv;
  const float* hb = hg + (size_t)bix * NIv;

#if HAVE_ASYNC_LDS
  // stage Q and G straight into LDS via async B128 copies (ASYNCcnt-tracked)
  for (int i = tid * 4; i < NXv * NXv; i += NTH * 4)
    __builtin_amdgcn_global_load_async_to_lds_b128((p1_v4i)(Qb + i), (p3_v4i)&sh.U[i], 0, 0);
  for (int i = tid * 4; i < NIv * NXv; i += NTH * 4)
    __builtin_amdgcn_global_load_async_to_lds_b128((p1_v4i)(Gb + i), (p3_v4i)&sh.G[i], 0, 0);
  __builtin_amdgcn_s_wait_asynccnt(0);
#else
  for (int i = tid; i < NXv * NXv; i += NTH) sh.U[i] = Qb[i];
  for (int i = tid; i < NIv * NXv; i += NTH) sh.G[i] = Gb[i];
#endif
  if (tid < NXv) { sh.A[tid] = Ab[tid]; sh.p[tid] = pb[tid]; }
  if (tid < NIv) sh.h[tid] = hb[tid];
  if (tid == 0) sh.b = bg[bix];
  __syncthreads();

  // ---- pre_factor_kkt ----
  chol_upper_blocked(sh.U, NXv, tid);
  if (tid < NXv) sh.idU[tid] = 1.0f / sh.U[tid * NXv + tid];
  __syncthreads();

  // invQ_GT = cho_solve(U_Q, G^T), 128 columns in parallel; stored in sh.S (aliased)
  float* IQ = sh.S;
  if (tid < NIv) {
    const int c = tid;
    for (int i = 0; i < NXv; ++i) {  // forward: U^T y = G[c,:]
      float acc = sh.G[c * NXv + i];
      for (int k = 0; k < i; ++k) acc -= sh.U[k * NXv + i] * IQ[k * NIv + c];
      IQ[i * NIv + c] = acc * sh.idU[i];
    }
    for (int i = NXv - 1; i >= 0; --i) {  // backward: U x = y
      float acc = IQ[i * NIv + c];
      for (int k = i + 1; k < NXv; ++k) acc -= sh.U[i * NXv + k] * IQ[k * NIv + c];
      IQ[i * NIv + c] = acc * sh.idU[i];
    }
  }
  __syncthreads();

  // ---- R = G(128x64) @ invQ_GT(64x128) via V_WMMA_F32_16X16X4_F32 ----
  {
    const int lane = tid & 31;
    const int wave = tid >> 5;
    const int m0 = wave * 16;      // 8 waves -> 8 row blocks of C
    const int hh = lane >> 4;      // lane half
    const int ll = lane & 15;      // M for A-frag, N for B/C-frag
#if HAVE_WMMA_F32
    for (int nb = 0; nb < 8; ++nb) {
      const int n0 = nb * 16;
      v8f acc = {0.f, 0.f, 0.f, 0.f, 0.f, 0.f, 0.f, 0.f};
      for (int k0 = 0; k0 < NXv; k0 += 4) {
        v2f af, bf;
        af.x = sh.G[(m0 + ll) * NXv + k0 + 2 * hh];
        af.y = sh.G[(m0 + ll) * NXv + k0 + 2 * hh + 1];
        bf.x = IQ[(k0 + 2 * hh) * NIv + n0 + ll];
        bf.y = IQ[(k0 + 2 * hh + 1) * NIv + n0 + ll];
        acc = __builtin_amdgcn_wmma_f32_16x16x4_f32(false, af, false, bf, (short)0, acc,
                                                    false, false);
      }
      for (int r = 0; r < 8; ++r) sh.R[(m0 + r + 8 * hh) * NIv + n0 + ll] = acc[r];
    }
#else
    for (int idx = tid; idx < NIv * NIv; idx += NTH) {
      int r = idx >> 7, c = idx & (NIv - 1);
      float acc = 0.0f;
      for (int k = 0; k < NXv; ++k) acc += sh.G[r * NXv + k] * IQ[k * NIv + c];
      sh.R[idx] = acc;
    }
    (void)m0; (void)hh; (void)ll;
#endif
    __syncthreads();
  }

  // invQ_AT, u11, U12, R -= U12^T U12
  for (int i = tid; i < NXv; i += NTH) sh.iqat[i] = sh.A[i];
  trsv_lt(sh.U, NXv, sh.iqat, sh.idU, tid);
  trsv_ut(sh.U, NXv, sh.iqat, sh.idU, tid);
  if (tid == 0) {
    float acc = 0.0f;
    for (int k = 0; k < NXv; ++k) acc += sh.A[k] * sh.iqat[k];
    sh.u11 = sqrtf(acc);
  }
  __syncthreads();
  if (tid < NIv) {
    float acc = 0.0f;
    for (int k = 0; k < NXv; ++k) acc += sh.G[tid * NXv + k] * sh.iqat[k];
    sh.U12[tid] = acc / sh.u11;
  }
  __syncthreads();
  for (int idx = tid; idx < NIv * NIv; idx += NTH) {
    int r = idx >> 7, c = idx & (NIv - 1);
    sh.R[idx] -= sh.U12[r] * sh.U12[c];
  }
  __syncthreads();

  // ---- initial factor + solve (d0 = 1) ----
  if (tid < NIv) sh.d[tid] = 1.0f;
  __syncthreads();
  factor_kkt(sh, tid);
  if (tid < NIv) sh.rzv[tid] = -sh.h[tid];
  __syncthreads();
  solve_kkt(sh, sh.p, nullptr, sh.rzv, -sh.b, sh.x, sh.s, sh.z, &sh.yv, tid);

  // shift s, z to positivity
  float mn_s = block_min(sh, (tid < NIv) ? sh.s[tid] : INF_F, tid);
  float mn_z = block_min(sh, (tid < NIv) ? sh.z[tid] : INF_F, tid);
  float shs = (1.0f - mn_s > 1.0f) ? (1.0f - mn_s) : 0.0f;
  float shz = (1.0f - mn_z > 1.0f) ? (1.0f - mn_z) : 0.0f;
  if (tid < NIv) { sh.s[tid] += shs; sh.z[tid] += shz; }
  __syncthreads();

  // ---- interior point iterations ----
  for (int it = 0; it < NITER; ++it) {
    // qx = U^T (U x) == Q x ; first t64 = U x
    if (tid < NXv) {
      float acc = 0.0f;
      for (int j = tid; j < NXv; ++j) acc += sh.U[tid * NXv + j] * sh.x[j];
      sh.t64[tid] = acc;
    }
    __syncthreads();
    if (tid < NXv) {
      float acc = sh.p[tid] + sh.A[tid] * sh.yv;
      for (int k = 0; k <= tid; ++k) acc += sh.U[k * NXv + tid] * sh.t64[k];
      for (int r = 0; r < NIv; ++r) acc += sh.G[r * NXv + tid] * sh.z[r];
      sh.rxv[tid] = acc;
    }
    if (tid < NIv) {
      float acc = sh.s[tid] - sh.h[tid];
      for (int k = 0; k < NXv; ++k) acc += sh.G[tid * NXv + k] * sh.x[k];
      sh.rzv[tid] = acc;
    }
    if (tid == 0) {
      float acc = -sh.b;
      for (int k = 0; k < NXv; ++k) acc += sh.A[k] * sh.x[k];
      sh.ryv = acc;
    }
    if (tid < NIv) sh.d[tid] = sh.z[tid] / sh.s[tid];
    __syncthreads();

    float szsum = block_sum(sh, (tid < NIv) ? sh.s[tid] * sh.z[tid] : 0.0f, tid);
    float mu = szsum / (float)NIv;

    factor_kkt(sh, tid);

    // affine (predictor) step; rs = z
    solve_kkt(sh, sh.rxv, sh.z, sh.rzv, sh.ryv, sh.dxa, sh.dsa, sh.dza, &sh.dya, tid);

    float a_z = step_min(sh, sh.z, sh.dza, tid);
    float a_s = step_min(sh, sh.s, sh.dsa, tid);
    float alpha = 0.999f * fminf(fminf(a_z, a_s), 1.0f);

    float spzp = block_sum(
        sh,
        (tid < NIv) ? (sh.s[tid] + alpha * sh.dsa[tid]) * (sh.z[tid] + alpha * sh.dza[tid])
                    : 0.0f,
        tid);
    float ratio = spzp / szsum;
    float sig = ratio * ratio * ratio;
    float musig = -mu * sig;
    if (tid < NIv) sh.t128b[tid] = (musig + sh.dsa[tid] * sh.dza[tid]) / sh.s[tid];
    __syncthreads();

    // corrector step
    solve_kkt(sh, nullptr, sh.t128b, nullptr, 0.0f, sh.dxc, sh.dsc, sh.dzc, &sh.dyc, tid);

    if (tid < NXv) sh.dxa[tid] += sh.dxc[tid];
    if (tid < NIv) { sh.dsa[tid] += sh.dsc[tid]; sh.dza[tid] += sh.dzc[tid]; }
    if (tid == 0) sh.dya += sh.dyc;
    __syncthreads();

    float a_z2 = step_min(sh, sh.z, sh.dza, tid);
    float a_s2 = step_min(sh, sh.s, sh.dsa, tid);
    float alpha2 = 0.999f * fminf(fminf(a_z2, a_s2), 1.0f);

    if (tid < NXv) sh.x[tid] += alpha2 * sh.dxa[tid];
    if (tid < NIv) { sh.s[tid] += alpha2 * sh.dsa[tid]; sh.z[tid] += alpha2 * sh.dza[tid]; }
    if (tid == 0) sh.yv += alpha2 * sh.dya;
    __syncthreads();
  }

  if (tid < NXv) out[(size_t)bix * NXv + tid] = sh.x[tid];
}

extern "C" void kernel_launch(void* const* d_in, const int* in_sizes, int n_in, void* d_out,
                              int out_size, void* d_ws, size_t ws_size, hipStream_t stream) {
  const float* Q = (const float*)d_in[0];
  const float* p = (const float*)d_in[1];
  const float* A = (const float*)d_in[2];
  const float* b = (const float*)d_in[3];
  const float* G = (const float*)d_in[4];
  const float* h = (const float*)d_in[5];
  float* out = (float*)d_out;
  const int Bn = in_sizes[0] / (NXv * NXv);
  (void)n_in; (void)out_size; (void)d_ws; (void)ws_size;
  optnet_ipm_kernel<<<dim3(Bn), dim3(NTH), 0, stream>>>(Q, p, A, b, G, h, out);
}